// VDWModule_66108136620165
// MI455X (gfx1250) — compile-verified
//
#include <hip/hip_runtime.h>

#define MOL 64
#define ATOM 512
#define TILE 16
#define NTILE (ATOM / TILE)                   // 32 tiles per axis
#define NPAIR_TILES (NTILE * (NTILE + 1) / 2) // 528 upper-tri tiles (incl diag)
#define THREADS 512
#define WAVES_PER_BLOCK (THREADS / 32)        // 16 wave32s
#define SPLIT 8                               // blocks per molecule (uses d_ws)

typedef float v2f __attribute__((ext_vector_type(2)));
typedef float v8f __attribute__((ext_vector_type(8)));

// One block handles (1/nsplit) of one molecule's 16x16 pair tiles.
// Per-pair dot products r_i . r_j come from V_WMMA_F32_16X16X4_F32 (Gram matrix,
// K=3 padded to 4); dist2 = |ri|^2 + |rj|^2 - 2*dot.
__global__ __launch_bounds__(THREADS)
void vdw_pair_kernel(const int* __restrict__ species,
                     const float* __restrict__ energies,
                     const float* __restrict__ c6,
                     const float* __restrict__ coords,
                     float* __restrict__ out_species_f,
                     float* __restrict__ partial,   // [MOL*nsplit], or out_energy if direct
                     int nsplit, int direct)
{
    __shared__ float sx[ATOM], sy[ATOM], sz[ATOM];
    __shared__ float sn[ATOM], sc6[ATOM], sval[ATOM];
    __shared__ float wsum[WAVES_PER_BLOCK];

    const int blk  = blockIdx.x;
    const int m    = blk / nsplit;
    const int s    = blk % nsplit;
    const int tid  = threadIdx.x;
    const int lane = tid & 31;
    const int wave = tid >> 5;

    // ---- per-atom precompute into LDS (512 threads <-> 512 atoms) ----
    {
        const int a = tid;
        const float* cb = coords + ((size_t)m * ATOM + a) * 3;
        float x = cb[0], y = cb[1], z = cb[2];
        sx[a] = x; sy[a] = y; sz[a] = z;
        sn[a] = x * x + y * y + z * z;
        float cv = c6[(size_t)m * ATOM + a];
        sc6[a] = 1000.0f / (1.0f + __expf(cv));     // 1000*sigmoid(-c6)
        int sp = species[(size_t)m * ATOM + a];
        sval[a] = (sp <= -1) ? 0.0f : 1.0f;
        if (s == 0) out_species_f[(size_t)m * ATOM + a] = (float)sp;
    }
    __syncthreads();

    const int  mrow = lane & 15;
    const bool hihalf = (lane >= 16);
    float acc = 0.0f;

    const int gw0      = s * WAVES_PER_BLOCK + wave;      // global wave id in molecule
    const int gwstride = nsplit * WAVES_PER_BLOCK;

    for (int t = gw0; t < NPAIR_TILES; t += gwstride) {
        // decode linear t -> (ti, tj) with ti <= tj (upper triangle incl diagonal)
        int ti = 0, rem = t;
        while (rem >= NTILE - ti) { rem -= NTILE - ti; ++ti; }
        const int tj = ti + rem;
        const int i0 = ti * TILE, j0 = tj * TILE;

        // A frag: 16x4 f32 (rows i0..i0+15). lanes0-15: K0=x,K1=y; lanes16-31: K2=z,K3=0
        // B frag: 4x16 f32 (cols j0..j0+15), mirrored layout.
        v2f af, bf;
        af.x = hihalf ? sz[i0 + mrow] : sx[i0 + mrow];
        af.y = hihalf ? 0.0f          : sy[i0 + mrow];
        bf.x = hihalf ? sz[j0 + mrow] : sx[j0 + mrow];
        bf.y = hihalf ? 0.0f          : sy[j0 + mrow];
        v8f cf = {0.f, 0.f, 0.f, 0.f, 0.f, 0.f, 0.f, 0.f};
        // (neg_a, A, neg_b, B, c_mod, C, reuse_a, reuse_b)
        v8f dotv = __builtin_amdgcn_wmma_f32_16x16x4_f32(
            false, af, false, bf, (short)0, cf, false, false);

        // D layout: lane l, VGPR v -> row (v + 8*(l>=16)), col (l&15)
        const int   jg = j0 + mrow;
        const float nj = sn[jg];
        const float vj = sval[jg];
        const float cj = sc6[jg];
#pragma unroll
        for (int v = 0; v < 8; ++v) {
            const int   ig  = i0 + v + (hihalf ? 8 : 0);
            const float ni  = sn[ig];
            const float d2  = ni + nj - 2.0f * dotv[v];
            const float d6  = d2 * d2 * d2;
            const float ci  = sc6[ig];
            const float cij = 2.0f * ci * cj / (ci + cj);       // harmonic mean
            const float fd  = 1.0f / (1.0f + __expf(25.0f - d2)); // sigmoid(d2-25)
            const float e   = -fd * cij / d6 * sval[ig] * vj;
            acc += (ig < jg) ? e : 0.0f;  // strict upper triangle (branchless)
        }
    }

    // ---- deterministic reduction: wave shuffle -> LDS -> thread 0 ----
#pragma unroll
    for (int off = 16; off > 0; off >>= 1)
        acc += __shfl_down(acc, off, 32);
    if (lane == 0) wsum[wave] = acc;
    __syncthreads();
    if (tid == 0) {
        float bs = 0.0f;
#pragma unroll
        for (int w = 0; w < WAVES_PER_BLOCK; ++w) bs += wsum[w];
        if (direct) partial[m] = energies[m] + bs;
        else        partial[(size_t)m * nsplit + s] = bs;
    }
}

__global__ void vdw_reduce_kernel(const float* __restrict__ energies,
                                  const float* __restrict__ partial,
                                  float* __restrict__ out_energy,
                                  int nsplit)
{
    int m = threadIdx.x;
    if (m < MOL) {
        float sum = 0.0f;
        for (int s = 0; s < nsplit; ++s) sum += partial[(size_t)m * nsplit + s];
        out_energy[m] = energies[m] + sum;
    }
}

extern "C" void kernel_launch(void* const* d_in, const int* in_sizes, int n_in,
                              void* d_out, int out_size, void* d_ws, size_t ws_size,
                              hipStream_t stream)
{
    const int*   species  = (const int*)d_in[0];
    const float* energies = (const float*)d_in[1];
    const float* c6       = (const float*)d_in[2];
    const float* coords   = (const float*)d_in[3];

    float* out           = (float*)d_out;
    float* out_species_f = out;                      // MOL*ATOM floats
    float* out_energy    = out + (size_t)MOL * ATOM; // MOL floats

    const size_t need = (size_t)MOL * SPLIT * sizeof(float);
    if (d_ws != nullptr && ws_size >= need) {
        float* partial = (float*)d_ws;
        vdw_pair_kernel<<<MOL * SPLIT, THREADS, 0, stream>>>(
            species, energies, c6, coords, out_species_f, partial, SPLIT, 0);
        vdw_reduce_kernel<<<1, MOL, 0, stream>>>(energies, partial, out_energy, SPLIT);
    } else {
        // Fallback: one block per molecule, writes energies directly (no scratch).
        vdw_pair_kernel<<<MOL, THREADS, 0, stream>>>(
            species, energies, c6, coords, out_species_f, out_energy, 1, 1);
    }
}